// HONU_5961414607272
// MI455X (gfx1250) — compile-verified
//
#include <hip/hip_runtime.h>

typedef __attribute__((ext_vector_type(2))) float v2f;
typedef __attribute__((ext_vector_type(4))) float v4f;
typedef __attribute__((ext_vector_type(8))) float v8f;

#define LIN 64
#define SA_STRIDE 68   // 64 + 4 pad: bank-conflict-free for K-strided v2f reads
#define SY_STRIDE 68

// ---------------------------------------------------------------------------
// Kernel 1: build AT[n][k] = (k<=n) ? W[k*64 - k(k-1)/2 + (n-k)] : 0
// (transposed upper-triangular quadratic-form matrix; 64x64 = 16 KB in d_ws)
// ---------------------------------------------------------------------------
__global__ void honu_build_AT(const float* __restrict__ W,
                              float* __restrict__ AT) {
    int t = blockIdx.x * blockDim.x + threadIdx.x;   // 0..4095
    int n = t >> 6;
    int k = t & 63;
    float v = 0.0f;
    if (k <= n) {
        int p = k * LIN - (k * (k - 1)) / 2 + (n - k);  // max 2079 < 2080
        v = W[p];
    }
    AT[t] = v;
}

// ---------------------------------------------------------------------------
// Kernel 2: out[i] = rowdot(X @ A, X) + b  via V_WMMA_F32_16X16X4_F32
// 4 waves / block, each wave owns a 16-row tile of X.
// ---------------------------------------------------------------------------
__global__ __launch_bounds__(128) void honu_wmma_f32(
    const float* __restrict__ x,     // (16384, 64)
    const float* __restrict__ AT,    // (64, 64) transposed A
    const float* __restrict__ bias,  // (1,)
    float* __restrict__ out) {       // (16384,)

    __shared__ float sA[LIN * SA_STRIDE];       // 17408 B, shared by block
    __shared__ float sY[4 * 16 * SY_STRIDE];    // 17408 B, per-wave Y tiles

    const int tid  = threadIdx.x;
    const int wave = tid >> 5;
    const int lane = tid & 31;
    const int half = lane >> 4;   // 0: K pair (0,1)  1: K pair (2,3)
    const int lrow = lane & 15;   // row (A-frag) / column (B-frag) index

    // Cooperative copy AT (row-major, stride 64) -> sA (stride 68), float4.
    for (int i4 = tid; i4 < 1024; i4 += 128) {
        int n = i4 >> 4;
        int k = (i4 & 15) << 2;
        *(v4f*)&sA[n * SA_STRIDE + k] = *(const v4f*)&AT[(n << 6) + k];
    }
    __syncthreads();

    const int tile = blockIdx.x * 4 + wave;   // 16-row tile index
    const int row  = tile * 16 + lrow;
    const float* xr = x + row * LIN;

    // Preload A-operand fragments: 16 K-steps of 4, v2f per lane per step.
    v2f a[16];
#pragma unroll
    for (int s = 0; s < 16; ++s)
        a[s] = *(const v2f*)&xr[s * 4 + half * 2];

    v8f acc[4] = {};

    // Y(16x64) = Xtile(16x64) @ A(64x64): 4 column blocks x 16 K-steps.
#pragma unroll
    for (int cb = 0; cb < 4; ++cb) {
        const float* bcol = &sA[(cb * 16 + lrow) * SA_STRIDE + half * 2];
#pragma unroll
        for (int s = 0; s < 16; ++s) {
            v2f bfrag = *(const v2f*)&bcol[s * 4];
            acc[cb] = __builtin_amdgcn_wmma_f32_16x16x4_f32(
                /*neg_a=*/false, a[s], /*neg_b=*/false, bfrag,
                /*c_mod=*/(short)0, acc[cb],
                /*reuse_a=*/false, /*reuse_b=*/false);
        }
    }

    // Spill Y (D layout: lane = column, VGPR v = row v + half*8) to LDS.
    float* y = &sY[wave * 16 * SY_STRIDE];
#pragma unroll
    for (int cb = 0; cb < 4; ++cb) {
        int col = cb * 16 + lrow;
#pragma unroll
        for (int v = 0; v < 8; ++v)
            y[(v + half * 8) * SY_STRIDE + col] = acc[cb][v];
    }
    __syncthreads();

    // out[i] = dot(Y[i,:], X[i,:]) + b ; lanes 0-15 each own one row.
    if (half == 0) {
        float sum = 0.0f;
#pragma unroll
        for (int c4 = 0; c4 < 16; ++c4) {
            v4f yv = *(const v4f*)&y[lrow * SY_STRIDE + c4 * 4];
            v4f xv = *(const v4f*)&xr[c4 * 4];
            sum += yv.x * xv.x + yv.y * xv.y + yv.z * xv.z + yv.w * xv.w;
        }
        out[row] = sum + bias[0];
    }
}

// ---------------------------------------------------------------------------
extern "C" void kernel_launch(void* const* d_in, const int* in_sizes, int n_in,
                              void* d_out, int out_size, void* d_ws, size_t ws_size,
                              hipStream_t stream) {
    const float* x = (const float*)d_in[0];   // (16384, 64) f32
    const float* W = (const float*)d_in[1];   // (2145,) f32 (first 2080 used)
    const float* b = (const float*)d_in[2];   // (1,) f32
    float* out = (float*)d_out;               // (16384,) f32
    float* AT  = (float*)d_ws;                // 64*64 f32 = 16 KB scratch

    honu_build_AT<<<16, 256, 0, stream>>>(W, AT);

    // 16384 rows / (4 waves * 16 rows) = 256 blocks of 128 threads.
    honu_wmma_f32<<<256, 128, 0, stream>>>(x, AT, b, out);
}